// ComplexAttention_31679678775908
// MI455X (gfx1250) — compile-verified
//
#include <hip/hip_runtime.h>

#define SEQ 2048
#define CH  512
#define NH  16
#define HD  32

typedef float v2f __attribute__((ext_vector_type(2)));
typedef float v8f __attribute__((ext_vector_type(8)));

__device__ __forceinline__ v8f wmma4(v2f a, v2f b, v8f c) {
  // V_WMMA_F32_16X16X4_F32 : D = A(16x4) * B(4x16) + C
  return __builtin_amdgcn_wmma_f32_16x16x4_f32(false, a, false, b, (short)0, c, false, false);
}

__device__ __forceinline__ v2f ldp(const float* p) {       // aligned pair load -> v2f
  const float2 t = *(const float2*)p;
  v2f r = { t.x, t.y };
  return r;
}

// ---------------------------------------------------------------------------
// Complex projection: Y[s,o] = sum_i X[s,i]*W[o,i] + b[o]
// X: [SEQ][CH][2] interleaved, W: [CH][CH][2] (out,in), b: [CH][2]
// MODE 0: planar output  Yre/Yim = [NH][SEQ][HD]          (q, k)
// MODE 1: transposed planar output Yre/Yim = [NH][HD][SEQ] (v)
// One wave per 16x16 complex output tile; 4 f32 WMMAs per K-step of 4.
// ---------------------------------------------------------------------------
template <int MODE>
__global__ void __launch_bounds__(256)
proj_complex_kernel(const float* __restrict__ X, const float* __restrict__ W,
                    const float* __restrict__ Bv, float* __restrict__ Yre,
                    float* __restrict__ Yim) {
  const int wave = threadIdx.x >> 5;
  const int lane = threadIdx.x & 31;
  const int half = lane >> 4;
  const int l16  = lane & 15;

  const int tileId  = blockIdx.x * 8 + wave;
  const int nTilesO = CH / 16;
  const int sTile   = tileId / nTilesO;
  const int oTile   = tileId % nTilesO;

  const int arow = sTile * 16 + l16;               // A row (m)
  const int brow = oTile * 16 + l16;               // B "row" (n = out channel)

  v8f yr = {0.f}, yi = {0.f};

  for (int k0 = 0; k0 < CH; k0 += 4) {
    const int ka = k0 + half * 2;
    const float4 xa = *(const float4*)(X + ((size_t)arow * CH + ka) * 2);
    const float4 wb = *(const float4*)(W + ((size_t)brow * CH + ka) * 2);
    v2f ar  = { xa.x,  xa.z };
    v2f ai  = { xa.y,  xa.w };
    v2f ain = { -xa.y, -xa.w };                    // -xi (one negation per A load)
    v2f br  = { wb.x,  wb.z };
    v2f bi  = { wb.y,  wb.w };
    yr = wmma4(ar,  br, yr);                       // + xr*wr
    yr = wmma4(ain, bi, yr);                       // - xi*wi
    yi = wmma4(ar,  bi, yi);                       // + xr*wi
    yi = wmma4(ai,  br, yi);                       // + xi*wr
  }

  const float2 bb = *(const float2*)(Bv + (size_t)brow * 2);
  const int h  = brow >> 5;
  const int dd = brow & 31;

  if (MODE == 0) {
    // planar [h][s][dd]
#pragma unroll
    for (int r = 0; r < 8; ++r) {                  // C/D: m = r + 8*half, n = l16
      const int s = sTile * 16 + r + 8 * half;
      const size_t idx = ((size_t)h * SEQ + s) * HD + dd;
      Yre[idx] = yr[r] + bb.x;
      Yim[idx] = yi[r] + bb.y;
    }
  } else {
    // transposed planar [h][dd][s]; each lane's 8 rows are consecutive s values
    const size_t base = ((size_t)(h * HD + dd)) * SEQ + sTile * 16 + 8 * half;
    float4 r0 = { yr[0] + bb.x, yr[1] + bb.x, yr[2] + bb.x, yr[3] + bb.x };
    float4 r1 = { yr[4] + bb.x, yr[5] + bb.x, yr[6] + bb.x, yr[7] + bb.x };
    float4 i0 = { yi[0] + bb.y, yi[1] + bb.y, yi[2] + bb.y, yi[3] + bb.y };
    float4 i1 = { yi[4] + bb.y, yi[5] + bb.y, yi[6] + bb.y, yi[7] + bb.y };
    *(float4*)(Yre + base)     = r0;
    *(float4*)(Yre + base + 4) = r1;
    *(float4*)(Yim + base)     = i0;
    *(float4*)(Yim + base + 4) = i1;
  }
}

// ---------------------------------------------------------------------------
// Flash attention over complex q,k,v.
// q,k planar [NH][SEQ][HD]; v transposed planar [NH][HD][SEQ], re/im planes.
// scores = q . conj(k) / sqrt(HD); weights = softmax(|scores|); out = w @ v
// One wave per 16-query tile; streaming over 16-key tiles with online softmax.
// All WMMA operands are direct aligned float2 loads (no packing movs/xors).
// ---------------------------------------------------------------------------
__global__ void __launch_bounds__(256)
attn_complex_kernel(const float* __restrict__ Qre, const float* __restrict__ Qim,
                    const float* __restrict__ Kre, const float* __restrict__ Kim,
                    const float* __restrict__ Vtre, const float* __restrict__ Vtim,
                    float* __restrict__ Out) {
  __shared__ float lds_p[8][16 * 16];              // per-wave C->A relayout scratch

  const int wave = threadIdx.x >> 5;
  const int lane = threadIdx.x & 31;
  const int half = lane >> 4;
  const int l16  = lane & 15;
  const int h    = blockIdx.y;
  const int q0   = (blockIdx.x * 8 + wave) * 16;

  const float scale = 0.17677669529663687f;        // 1/sqrt(32)

  const float* qreh = Qre + (size_t)h * SEQ * HD;
  const float* qimh = Qim + (size_t)h * SEQ * HD;
  const float* kreh = Kre + (size_t)h * SEQ * HD;
  const float* kimh = Kim + (size_t)h * SEQ * HD;
  const float* vreh = Vtre + (size_t)h * HD * SEQ;
  const float* vimh = Vtim + (size_t)h * HD * SEQ;
  float* pbuf = lds_p[wave];

  // Preload q A-tile: 8 K-steps of 4; also pre-negate qr once (conj(k) trick:
  // sim = qi*kr + (-qr)*ki) so the 128-iteration key loop has zero negations.
  v2f qr[8], qi[8], qrn[8];
#pragma unroll
  for (int t = 0; t < 8; ++t) {
    const int dk = t * 4 + half * 2;
    const size_t off = ((size_t)(q0 + l16)) * HD + dk;
    qr[t]  = ldp(qreh + off);
    qi[t]  = ldp(qimh + off);
    qrn[t] = -qr[t];
  }

  float m_run[8], l_run[8];
  v8f o_re0 = {0.f}, o_re1 = {0.f}, o_im0 = {0.f}, o_im1 = {0.f};
#pragma unroll
  for (int r = 0; r < 8; ++r) { m_run[r] = -1e30f; l_run[r] = 0.f; }

  for (int kt = 0; kt < SEQ / 16; ++kt) {
    const int kbase = kt * 16;

    // ---- score tile: S = q . conj(k)  (complex), 32 WMMAs ----
    v8f sre = {0.f}, sim = {0.f};
#pragma unroll
    for (int t = 0; t < 8; ++t) {
      const int dk = t * 4 + half * 2;
      const size_t off = ((size_t)(kbase + l16)) * HD + dk;
      v2f krv = ldp(kreh + off);
      v2f kiv = ldp(kimh + off);
      sre = wmma4(qr[t],  krv, sre);               // + qr*kr
      sre = wmma4(qi[t],  kiv, sre);               // + qi*ki   (conj)
      sim = wmma4(qi[t],  krv, sim);               // + qi*kr
      sim = wmma4(qrn[t], kiv, sim);               // - qr*ki   (conj)
    }

    // ---- |s|*scale, online softmax (rows live in 16-lane halves) ----
    float pl[8], rs[8];
#pragma unroll
    for (int r = 0; r < 8; ++r) {
      float a = sqrtf(sre[r] * sre[r] + sim[r] * sim[r]) * scale;
      float mx = a;
#pragma unroll
      for (int msk = 1; msk < 16; msk <<= 1) mx = fmaxf(mx, __shfl_xor(mx, msk, 32));
      const float newm = fmaxf(m_run[r], mx);
      float pv = __expf(a - newm);
      float sum = pv;
#pragma unroll
      for (int msk = 1; msk < 16; msk <<= 1) sum += __shfl_xor(sum, msk, 32);
      const float sc = __expf(m_run[r] - newm);
      l_run[r] = l_run[r] * sc + sum;
      m_run[r] = newm;
      pl[r] = pv;
      rs[r] = sc;
    }
#pragma unroll
    for (int r = 0; r < 8; ++r) {
      o_re0[r] *= rs[r]; o_re1[r] *= rs[r];
      o_im0[r] *= rs[r]; o_im1[r] *= rs[r];
    }

    // ---- relayout p (C-layout -> A-layout) through per-wave LDS ----
#pragma unroll
    for (int r = 0; r < 8; ++r) pbuf[(r + 8 * half) * 16 + l16] = pl[r];
    __asm__ volatile("s_wait_dscnt 0x0" ::: "memory");

    // ---- out += p @ v  (complex), K = 16 keys -> 4 steps, 16 WMMAs ----
#pragma unroll
    for (int t = 0; t < 4; ++t) {
      const int klocal = t * 4 + half * 2;
      v2f aP = ldp(pbuf + l16 * 16 + klocal);
      const size_t voff = (size_t)l16 * SEQ + kbase + klocal;   // [dd][s] pair
      v2f br0 = ldp(vreh + voff);                  // dd tile 0 (n = l16)
      v2f bi0 = ldp(vimh + voff);
      v2f br1 = ldp(vreh + voff + (size_t)16 * SEQ);  // dd tile 1 (n = 16+l16)
      v2f bi1 = ldp(vimh + voff + (size_t)16 * SEQ);
      o_re0 = wmma4(aP, br0, o_re0);
      o_im0 = wmma4(aP, bi0, o_im0);
      o_re1 = wmma4(aP, br1, o_re1);
      o_im1 = wmma4(aP, bi1, o_im1);
    }
  }

  // ---- normalize and store interleaved [SEQ][512][2] ----
#pragma unroll
  for (int r = 0; r < 8; ++r) {
    const float inv = 1.0f / l_run[r];
    const int s  = q0 + r + 8 * half;
    const int o0 = h * HD + l16;
    float2 e0 = { o_re0[r] * inv, o_im0[r] * inv };
    float2 e1 = { o_re1[r] * inv, o_im1[r] * inv };
    *(float2*)(Out + ((size_t)s * CH + o0) * 2)      = e0;
    *(float2*)(Out + ((size_t)s * CH + o0 + 16) * 2) = e1;
  }
}

extern "C" void kernel_launch(void* const* d_in, const int* in_sizes, int n_in,
                              void* d_out, int out_size, void* d_ws, size_t ws_size,
                              hipStream_t stream) {
  (void)in_sizes; (void)n_in; (void)out_size; (void)ws_size;
  // setup_inputs order: Q, V, K, Wq, bq, Wk, bk, Wv, bv
  const float* Q  = (const float*)d_in[0];
  const float* V  = (const float*)d_in[1];
  const float* K  = (const float*)d_in[2];
  const float* Wq = (const float*)d_in[3];
  const float* bq = (const float*)d_in[4];
  const float* Wk = (const float*)d_in[5];
  const float* bk = (const float*)d_in[6];
  const float* Wv = (const float*)d_in[7];
  const float* bv = (const float*)d_in[8];
  float* out = (float*)d_out;

  const size_t plane = (size_t)NH * SEQ * HD;      // 1M floats = 4 MB per plane
  float* qre = (float*)d_ws;
  float* qim = qre + plane;
  float* kre = qim + plane;
  float* kim = kre + plane;
  float* vre = kim + plane;                        // transposed planar
  float* vim = vre + plane;                        // total 24 MB of d_ws

  const dim3 blk(256);
  const dim3 pgrid((SEQ / 16) * (CH / 16) / 8);    // 512 blocks
  proj_complex_kernel<0><<<pgrid, blk, 0, stream>>>(Q, Wq, bq, qre, qim);
  proj_complex_kernel<0><<<pgrid, blk, 0, stream>>>(K, Wk, bk, kre, kim);
  proj_complex_kernel<1><<<pgrid, blk, 0, stream>>>(V, Wv, bv, vre, vim);

  const dim3 agrid(SEQ / 16 / 8, NH);              // 16 x 16 blocks
  attn_complex_kernel<<<agrid, blk, 0, stream>>>(qre, qim, kre, kim, vre, vim, out);
}